// RROIAlign_833223655745
// MI455X (gfx1250) — compile-verified
//
#include <hip/hip_runtime.h>
#include <cstdint>

// ---------------------------------------------------------------------------
// Rotated ROI Align fp32.  B=4, C=256, H=W=168, N=2000, 7x7 bins.
// Memory-bound gather: ~110MB feature reads (L2-resident on MI455X's 192MB L2)
// + ~100MB coalesced output writes -> ~9us HBM roofline.  No matmul structure
// (gathered operands vary per bin AND channel) so WMMA does not apply; CDNA5
// paths used: wave32 blocks + async global->LDS staging (ASYNCcnt).
// ---------------------------------------------------------------------------

#define RR_C      256
#define RR_H      168
#define RR_W      168
#define RR_HW     (RR_H * RR_W)
#define RR_PH     7
#define RR_PW     7
#define RR_NBIN   49
#define RR_SCALE  0.25f

#if defined(__HIP_DEVICE_COMPILE__) && defined(__gfx1250__)
  #if __has_builtin(__builtin_amdgcn_global_load_async_to_lds_b32)
    #define RR_ASYNC_LDS_B32(gp, lp)                                          \
      __builtin_amdgcn_global_load_async_to_lds_b32(                          \
          (__attribute__((address_space(1))) int*)(void*)(gp),                \
          (__attribute__((address_space(3))) int*)(void*)(lp), 0, 0)
  #else
    #define RR_ASYNC_LDS_B32(gp, lp)                                          \
      do {                                                                    \
        unsigned _lds = (unsigned)(uintptr_t)(lp);                            \
        asm volatile("global_load_async_to_lds_b32 %0, %1, off"               \
                     :: "v"(_lds), "v"((const void*)(gp)) : "memory");        \
      } while (0)
  #endif
  #if __has_builtin(__builtin_amdgcn_s_wait_asynccnt)
    #define RR_WAIT_ASYNC() __builtin_amdgcn_s_wait_asynccnt(0)
  #else
    #define RR_WAIT_ASYNC() asm volatile("s_wait_asynccnt 0" ::: "memory")
  #endif
#else
  #define RR_ASYNC_LDS_B32(gp, lp) (*(volatile float*)(lp) = *(const float*)(gp))
  #define RR_WAIT_ASYNC() ((void)0)
#endif

__device__ __forceinline__ float rr_round_half_away(float x) {
    // sign(x) * floor(|x| + 0.5); copysign reproduces sign(0)*... == 0 case.
    return copysignf(floorf(fabsf(x) + 0.5f), x);
}

__global__ __launch_bounds__(256) void rroi_align_kernel(
    const float* __restrict__ feat,   // (4, 256, 168, 168)
    const float* __restrict__ rois,   // (N, 6): bidx, cx, cy, w, h, angle
    float* __restrict__ out)          // (N, 256, 7, 7)
{
    __shared__ float  s_roi[8];          // staged ROI descriptor
    __shared__ int4   s_off[RR_NBIN];    // clamped gather offsets (lt, rt, lb, rb)
    __shared__ float4 s_w[RR_NBIN];      // validity-masked bilinear weights
    __shared__ int    s_plane;           // bidx * C * H * W

    const int n   = blockIdx.x;
    const int tid = threadIdx.x;

    // ---- stage this ROI's 6 floats into LDS via async DMA (gfx1250 path) ----
    if (tid < 6) {
        RR_ASYNC_LDS_B32(rois + (size_t)n * 6 + tid, &s_roi[tid]);
    }
    RR_WAIT_ASYNC();
    __syncthreads();

    // ---- per-bin geometry (threads 0..48) ----
    if (tid < RR_NBIN) {
        const float cx  = s_roi[1], cy = s_roi[2];
        const float rw  = s_roi[3], rh = s_roi[4];
        const float ang = s_roi[5];
        if (tid == 0) s_plane = ((int)s_roi[0]) * RR_C * RR_HW;

        const float angle = ang * 0.017453292519943295f; // pi/180
        float sa, ca;
        sincosf(angle, &sa, &ca);
        const float Sx = rw * RR_SCALE * (1.0f / RR_PW);
        const float Sy = rh * RR_SCALE * (1.0f / RR_PH);
        const float M00 = ca * Sx;
        const float M01 = sa * Sy;
        const float M02 = M00 * (-RR_PW * 0.5f) + M01 * (-RR_PH * 0.5f) + cx * RR_SCALE;
        const float M10 = -sa * Sx;
        const float M11 = ca * Sy;
        const float M12 = M10 * (-RR_PW * 0.5f) + M11 * (-RR_PH * 0.5f) + cy * RR_SCALE;

        const int ph = tid / RR_PW;
        const int pw = tid - ph * RR_PW;
        const float fx = (float)pw, fy = (float)ph;

        const float x00 = M00 * fx + M01 * fy + M02;
        const float y00 = M10 * fx + M11 * fy + M12;
        const float x01 = M00 * fx + M01 * (fy + 1.0f) + M02;
        const float y01 = M10 * fx + M11 * (fy + 1.0f) + M12;
        const float x10 = M00 * (fx + 1.0f) + M01 * fy + M02;
        const float y10 = M10 * (fx + 1.0f) + M11 * fy + M12;
        const float x11 = M00 * (fx + 1.0f) + M01 * (fy + 1.0f) + M02;
        const float y11 = M10 * (fx + 1.0f) + M11 * (fy + 1.0f) + M12;

        const float left  = fmaxf(rr_round_half_away(fminf(fminf(x00, x01), fminf(x10, x11))), 0.0f);
        const float right = fminf(rr_round_half_away(fmaxf(fmaxf(x00, x01), fmaxf(x10, x11))), (float)(RR_W - 1));
        const float top   = fmaxf(rr_round_half_away(fminf(fminf(y00, y01), fminf(y10, y11))), 0.0f);
        const float bot   = fminf(rr_round_half_away(fmaxf(fmaxf(y00, y01), fmaxf(y10, y11))), (float)(RR_H - 1));

        const float bcx = (left + right) * 0.5f;
        const float bcy = (top + bot) * 0.5f;
        const float xlf = floorf(bcx), xrf = ceilf(bcx);
        const float ytf = floorf(bcy), ybf = ceilf(bcy);
        const float rx = bcx - xlf, ry = bcy - ytf;

        const int xl = (int)xlf, xr = (int)xrf;
        const int yt = (int)ytf, yb = (int)ybf;
        const int xlc = min(max(xl, 0), RR_W - 1);
        const int xrc = min(max(xr, 0), RR_W - 1);
        const int ytc = min(max(yt, 0), RR_H - 1);
        const int ybc = min(max(yb, 0), RR_H - 1);

        // strict >0 validity, matching the reference
        const bool v_lt = (yt > 0) && (xl > 0) && (yt < RR_H) && (xl < RR_W);
        const bool v_rt = (yt > 0) && (xr > 0) && (yt < RR_H) && (xr < RR_W);
        const bool v_lb = (yb > 0) && (xl > 0) && (yb < RR_H) && (xl < RR_W);
        const bool v_rb = (yb > 0) && (xr > 0) && (yb < RR_H) && (xr < RR_W);

        const float omx = 1.0f - rx, omy = 1.0f - ry;
        s_off[tid] = make_int4(ytc * RR_W + xlc,   // lt
                               ytc * RR_W + xrc,   // rt
                               ybc * RR_W + xlc,   // lb
                               ybc * RR_W + xrc);  // rb
        s_w[tid] = make_float4(v_lt ? omx * omy : 0.0f,
                               v_rt ? rx  * omy : 0.0f,
                               v_lb ? omx * ry  : 0.0f,
                               v_rb ? rx  * ry  : 0.0f);
    }
    __syncthreads();

    // ---- main loop: idx = c*49 + bin, coalesced output stores ----
    const int plane = s_plane;
    float* __restrict__ outn = out + (size_t)n * (RR_C * RR_NBIN);

    #pragma unroll 1
    for (int it = 0; it < RR_NBIN; ++it) {
        const int idx = it * 256 + tid;                 // 0 .. 12543
        const unsigned c   = (unsigned)idx / RR_NBIN;
        const unsigned bin = (unsigned)idx - c * RR_NBIN;

        const float* __restrict__ p = feat + plane + c * RR_HW;
        const int4   o = s_off[bin];
        const float4 w = s_w[bin];

        const float acc = w.x * p[o.x] + w.y * p[o.y] + w.z * p[o.z] + w.w * p[o.w];
        outn[idx] = acc;
    }
}

extern "C" void kernel_launch(void* const* d_in, const int* in_sizes, int n_in,
                              void* d_out, int out_size, void* d_ws, size_t ws_size,
                              hipStream_t stream) {
    const float* feat = (const float*)d_in[0];
    const float* rois = (const float*)d_in[1];
    float* out = (float*)d_out;
    const int N = in_sizes[1] / 6;   // 2000 ROIs
    rroi_align_kernel<<<N, 256, 0, stream>>>(feat, rois, out);
}